// YoloLoss_79276506349681
// MI455X (gfx1250) — compile-verified
//
#include <hip/hip_runtime.h>
#include <hip/hip_bf16.h>
#include <math.h>

#define S_GRID 7
#define NB 2
#define NC 20
#define PRED_F (NB * 5 + NC)   // 30 floats per cell
#define TGT_F  (5 + NC)        // 25 floats per cell
#define L_OBJ  5.0f
#define L_NOBJ 0.5f
#define EPSV   1e-6f
#define CELLS_PER_BLOCK 256

typedef __attribute__((ext_vector_type(2))) float v2f;
typedef __attribute__((ext_vector_type(8))) float v8f;

__device__ __forceinline__ float sigf(float x) { return 1.0f / (1.0f + expf(-x)); }
__device__ __forceinline__ float reluf(float x) { return x > 0.0f ? x : 0.0f; }

__device__ __forceinline__ float iou_f(float cx1, float cy1, float w1, float h1,
                                       float cx2, float cy2, float w2, float h2) {
    float b1x1 = cx1 - w1 * 0.5f, b1y1 = cy1 - h1 * 0.5f;
    float b1x2 = cx1 + w1 * 0.5f, b1y2 = cy1 + h1 * 0.5f;
    float b2x1 = cx2 - w2 * 0.5f, b2y1 = cy2 - h2 * 0.5f;
    float b2x2 = cx2 + w2 * 0.5f, b2y2 = cy2 + h2 * 0.5f;
    float iw = fminf(b1x2, b2x2) - fmaxf(b1x1, b2x1); iw = fmaxf(iw, 0.0f);
    float ih = fminf(b1y2, b2y2) - fmaxf(b1y1, b2y1); ih = fmaxf(ih, 0.0f);
    float inter = iw * ih;
    float a1 = (b1x2 - b1x1) * (b1y2 - b1y1);
    float a2 = (b2x2 - b2x1) * (b2y2 - b2y1);
    return inter / (a1 + a2 - inter + EPSV);
}

__global__ void yolo_zero_out(float* out) { out[0] = 0.0f; }

__global__ __launch_bounds__(CELLS_PER_BLOCK)
void yolo_loss_kernel(const float* __restrict__ pred,
                      const float* __restrict__ target,
                      float* __restrict__ out,
                      int numCells, float invN) {
    // LDS staging tiles: 256 cells per block
    __shared__ float4 sPred4[CELLS_PER_BLOCK * PRED_F / 4]; // 1920 * 16B = 30720B
    __shared__ float4 sTgt4[(CELLS_PER_BLOCK * TGT_F + 3) / 4]; // 1600 * 16B = 25600B
    __shared__ float blockSum;

    const int tid = threadIdx.x;
    if (tid == 0) blockSum = 0.0f;

    const long long cellBase = (long long)blockIdx.x * CELLS_PER_BLOCK;

    // --- cooperative coalesced float4 tile loads (global_load_b128) ---
    {
        const float4* gp4 = reinterpret_cast<const float4*>(pred + cellBase * PRED_F);
        const long long totP4 = ((long long)numCells * PRED_F) / 4 - cellBase * PRED_F / 4;
        #pragma unroll
        for (int k = tid; k < CELLS_PER_BLOCK * PRED_F / 4; k += CELLS_PER_BLOCK)
            sPred4[k] = (k < totP4) ? gp4[k] : make_float4(0.f, 0.f, 0.f, 0.f);

        const float4* gt4 = reinterpret_cast<const float4*>(target + cellBase * TGT_F);
        const long long totT4 = ((long long)numCells * TGT_F) / 4 - cellBase * TGT_F / 4;
        #pragma unroll
        for (int k = tid; k < CELLS_PER_BLOCK * TGT_F / 4; k += CELLS_PER_BLOCK)
            sTgt4[k] = (k < totT4) ? gt4[k] : make_float4(0.f, 0.f, 0.f, 0.f);
    }
    __syncthreads();

    const float* sPred = reinterpret_cast<const float*>(sPred4);
    const float* sTgt  = reinterpret_cast<const float*>(sTgt4);

    // --- per-cell loss ---
    float partial = 0.0f;
    const long long g = cellBase + tid;
    if (g < numCells) {
        const float* P = sPred + tid * PRED_F;
        const float* T = sTgt + tid * TGT_F;

        // decode cell position: g = ((n*S + i)*S + j)
        int rem = (int)(g % (S_GRID * S_GRID));
        float fi = (float)(rem / S_GRID);
        float fj = (float)(rem % S_GRID);
        const float invS = 1.0f / (float)S_GRID;

        float tconf = T[0], tx = T[1], ty = T[2], tw = T[3], th = T[4];
        float t_x = (fj + tx) * invS;
        float t_y = (fi + ty) * invS;
        float twr = reluf(tw), thr = reluf(th);

        float iou_o[NB], iou_n[NB];
        #pragma unroll
        for (int b = 0; b < NB; ++b) {
            float px = (fj + P[b * 5 + 1]) * invS;
            float py = (fi + P[b * 5 + 2]) * invS;
            float pw = reluf(P[b * 5 + 3]);
            float ph = reluf(P[b * 5 + 4]);
            iou_o[b] = iou_f(px, py, pw, ph, t_x, t_y, tw, th);
            iou_n[b] = iou_f(px, py, pw, ph, t_x, t_y, twr, thr);
        }
        // argmax with first-max tie-break (jnp.argmax): pick 1 only if strictly greater
        int bo = (iou_o[1] > iou_o[0]) ? 1 : 0;
        int bn = (iou_n[1] > iou_n[0]) ? 1 : 0;

        // objectness branch values
        float bx = P[bo * 5 + 1], by = P[bo * 5 + 2];
        float bw = reluf(P[bo * 5 + 3]), bh = reluf(P[bo * 5 + 4]);
        float bconf_o = P[bo * 5 + 0];
        float biou = iou_o[bo];

        float dx = bx - tx, dy = by - ty;
        float xy_loss = dx * dx + dy * dy;
        float sw = sqrtf(fabsf(bw + EPSV)) - sqrtf(fabsf(tw + EPSV));
        float sh = sqrtf(fabsf(bh + EPSV)) - sqrtf(fabsf(th + EPSV));
        float wh_loss = sw * sw + sh * sh;
        float dco = sigf(bconf_o) - biou;
        float conf_obj = dco * dco;

        // softmax class loss
        float m = P[NB * 5];
        #pragma unroll
        for (int k = 1; k < NC; ++k) m = fmaxf(m, P[NB * 5 + k]);
        float se = 0.0f;
        float ex[NC];
        #pragma unroll
        for (int k = 0; k < NC; ++k) { ex[k] = expf(P[NB * 5 + k] - m); se += ex[k]; }
        float inv_se = 1.0f / se;
        float class_loss = 0.0f;
        #pragma unroll
        for (int k = 0; k < NC; ++k) {
            float d = ex[k] * inv_se - T[5 + k];
            class_loss += d * d;
        }

        float loss_obj = L_OBJ * (xy_loss + wh_loss) + conf_obj + class_loss;
        float sn = sigf(P[bn * 5 + 0]);
        float loss_noobj = L_NOBJ * sn * sn;

        float cell = (tconf == 1.0f) ? loss_obj : loss_noobj;
        partial = cell * invN;
    }

    // --- wave-level reduction via V_WMMA_F32_16X16X4_F32 (D = ones(16x4) x B + 0) ---
    // B VGPR0 carries the 32 per-lane partials (one (k,n) slot each), VGPR1 = 0.
    // D row 0 (c[0], lanes 0..15) = 16 column sums whose total is the wave sum.
    // Executed by all lanes (EXEC all-ones), outside any divergent branch.
    v2f a; a[0] = 1.0f; a[1] = 1.0f;
    v2f b; b[0] = partial; b[1] = 0.0f;
    v8f c = {};
    c = __builtin_amdgcn_wmma_f32_16x16x4_f32(
        /*neg_a=*/false, a, /*neg_b=*/false, b,
        /*c_mod=*/(short)0, c, /*reuse_a=*/false, /*reuse_b=*/false);
    float colsum = c[0];
    // fold the 16 column sums within each 16-lane half (halves hold duplicates)
    #pragma unroll
    for (int off = 1; off < 16; off <<= 1)
        colsum += __shfl_xor(colsum, off, 32);

    if ((tid & 31) == 0) atomicAdd(&blockSum, colsum);
    __syncthreads();
    if (tid == 0) atomicAdd(out, blockSum);
}

extern "C" void kernel_launch(void* const* d_in, const int* in_sizes, int n_in,
                              void* d_out, int out_size, void* d_ws, size_t ws_size,
                              hipStream_t stream) {
    const float* pred = (const float*)d_in[0];
    const float* target = (const float*)d_in[1];
    float* out = (float*)d_out;

    int N = in_sizes[0] / (S_GRID * S_GRID * PRED_F); // 16384
    int numCells = N * S_GRID * S_GRID;               // 802816
    int blocks = (numCells + CELLS_PER_BLOCK - 1) / CELLS_PER_BLOCK;

    yolo_zero_out<<<1, 1, 0, stream>>>(out);
    yolo_loss_kernel<<<blocks, CELLS_PER_BLOCK, 0, stream>>>(
        pred, target, out, numCells, 1.0f / (float)N);
}